// SEBlock_9070970929334
// MI455X (gfx1250) — compile-verified
//
#include <hip/hip_runtime.h>
#include <math.h>

typedef float v2f __attribute__((ext_vector_type(2)));
typedef float v4f __attribute__((ext_vector_type(4)));
typedef float v8f __attribute__((ext_vector_type(8)));

#define BATCH 32
#define CCH   256
#define CRED  64
#define HWTOT 3136   // 56*56
#define HW4   784    // HWTOT/4

// ---------------------------------------------------------------------------
// Pass 1: global average pool. One 256-thread block per (b,c) pair.
// b128 streaming loads (regular-temporal: warms L2 for pass 3),
// wave32 shuffle reduction + tiny LDS cross-wave combine.
// ---------------------------------------------------------------------------
__global__ void se_pool(const float* __restrict__ x, float* __restrict__ s) {
  __shared__ float red[8];
  const int bc = blockIdx.x;                       // b*256 + c
  const v4f* xp = (const v4f*)(x + (size_t)bc * HWTOT);
  float acc = 0.0f;
  for (int i = threadIdx.x; i < HW4; i += 256) {
    v4f v = xp[i];
    acc += v.x + v.y + v.z + v.w;
  }
  // wave32 reduce
  #pragma unroll
  for (int off = 16; off > 0; off >>= 1)
    acc += __shfl_down(acc, off, 32);
  const int wave = threadIdx.x >> 5;
  const int lane = threadIdx.x & 31;
  if (lane == 0) red[wave] = acc;
  __syncthreads();
  if (threadIdx.x == 0) {
    float t = 0.0f;
    #pragma unroll
    for (int w = 0; w < 8; ++w) t += red[w];
    s[bc] = t * (1.0f / (float)HWTOT);
  }
}

// ---------------------------------------------------------------------------
// Pass 2: SE MLP via V_WMMA_F32_16X16X4_F32. Single block, 8 waves (wave32).
//   h = relu(s @ w1^T + b1)   : M=32, K=256, N=64   (8 16x16 tiles, 1/wave)
//   e = h @ w2^T + b2         : M=32, K=64,  N=256  (32 tiles, 4/wave)
//   gate = sigmoid(e)
// f32 WMMA fragment layouts per CDNA5 ISA 7.12.2:
//   A 16x4 : lane<16 -> {K=k0, k0+1}, lane>=16 -> {K=k0+2, k0+3}, row = lane%16
//   B 4x16 : same K split per lane half, col = lane%16
//   C/D    : vgpr r, lane<16 -> row r,  lane>=16 -> row r+8, col = lane%16
// ---------------------------------------------------------------------------
__global__ void se_mlp(const float* __restrict__ s,    // [32,256] pooled means
                       const float* __restrict__ w1,   // [64,256] row-major
                       const float* __restrict__ b1,   // [64]
                       const float* __restrict__ w2,   // [256,64] row-major
                       const float* __restrict__ b2,   // [256]
                       float* __restrict__ gate)       // [32,256]
{
  __shared__ float s_lds[BATCH * CCH];   // 32 KB
  __shared__ float h_lds[BATCH * CRED];  // 8 KB

  for (int i = threadIdx.x; i < BATCH * CCH; i += 256) s_lds[i] = s[i];
  __syncthreads();

  const int wave = threadIdx.x >> 5;
  const int lane = threadIdx.x & 31;
  const int half = lane >> 4;      // 0: lanes 0-15, 1: lanes 16-31
  const int l16  = lane & 15;

  // ---- GEMM1: h = relu(s @ w1^T + b1) --------------------------------------
  {
    const int mt = wave & 1;       // M tile 0..1
    const int nt = wave >> 1;      // N tile 0..3
    const int m  = mt * 16 + l16;  // A row for this lane
    const int n  = nt * 16 + l16;  // B col for this lane
    v8f acc = {};
    for (int k0 = 0; k0 < CCH; k0 += 4) {
      const int ka = k0 + half * 2;
      v2f a, b;
      a.x = s_lds[m * CCH + ka];
      a.y = s_lds[m * CCH + ka + 1];
      b.x = w1[n * CCH + ka];          // w1^T[k][n] = w1[n][k]
      b.y = w1[n * CCH + ka + 1];
      acc = __builtin_amdgcn_wmma_f32_16x16x4_f32(
          false, a, false, b, (short)0, acc, false, false);
    }
#pragma unroll
    for (int r = 0; r < 8; ++r) {
      const int row = mt * 16 + (half ? r + 8 : r);
      const int col = nt * 16 + l16;
      float v = acc[r] + b1[col];
      h_lds[row * CRED + col] = v > 0.0f ? v : 0.0f;
    }
  }
  __syncthreads();

  // ---- GEMM2: gate = sigmoid(h @ w2^T + b2) --------------------------------
  for (int t = 0; t < 4; ++t) {
    const int tid = wave * 4 + t;  // 0..31 tiles
    const int mt = tid & 1;
    const int nt = tid >> 1;       // 0..15
    const int m  = mt * 16 + l16;
    const int n  = nt * 16 + l16;
    v8f acc = {};
    for (int k0 = 0; k0 < CRED; k0 += 4) {
      const int ka = k0 + half * 2;
      v2f a, b;
      a.x = h_lds[m * CRED + ka];
      a.y = h_lds[m * CRED + ka + 1];
      b.x = w2[n * CRED + ka];         // w2^T[k][n] = w2[n][k]
      b.y = w2[n * CRED + ka + 1];
      acc = __builtin_amdgcn_wmma_f32_16x16x4_f32(
          false, a, false, b, (short)0, acc, false, false);
    }
#pragma unroll
    for (int r = 0; r < 8; ++r) {
      const int row = mt * 16 + (half ? r + 8 : r);
      const int col = nt * 16 + l16;
      float e = acc[r] + b2[col];
      gate[row * CCH + col] = 1.0f / (1.0f + expf(-e));
    }
  }
}

// ---------------------------------------------------------------------------
// Pass 3: out = gate[b,c] * x. One block per (b,c); gate scalar broadcast.
// x reads are regular-temporal (should hit L2, warmed by pass 1).
// out stores are NON-TEMPORAL (TH=NT): out is never re-read, so don't let
// its write-allocation evict still-unread x lines from L2.
// ---------------------------------------------------------------------------
__global__ void se_scale(const float* __restrict__ x,
                         const float* __restrict__ gate,
                         float* __restrict__ out) {
  const int bc = blockIdx.x;
  const float g = gate[bc];
  const v4f* xp = (const v4f*)(x + (size_t)bc * HWTOT);
  v4f* op = (v4f*)(out + (size_t)bc * HWTOT);
  for (int i = threadIdx.x; i < HW4; i += 256) {
    v4f v = xp[i];
    v.x *= g; v.y *= g; v.z *= g; v.w *= g;
    __builtin_nontemporal_store(v, &op[i]);
  }
}

// ---------------------------------------------------------------------------
extern "C" void kernel_launch(void* const* d_in, const int* in_sizes, int n_in,
                              void* d_out, int out_size, void* d_ws, size_t ws_size,
                              hipStream_t stream) {
  const float* x  = (const float*)d_in[0];   // [32,256,56,56]
  const float* w1 = (const float*)d_in[1];   // [64,256]
  const float* b1 = (const float*)d_in[2];   // [64]
  const float* w2 = (const float*)d_in[3];   // [256,64]
  const float* b2 = (const float*)d_in[4];   // [256]
  float* out = (float*)d_out;

  float* s_ws    = (float*)d_ws;             // [32*256] pooled means
  float* gate_ws = s_ws + BATCH * CCH;       // [32*256] sigmoid gates

  const int nbc = BATCH * CCH;               // 8192 (b,c) pairs

  se_pool<<<nbc, 256, 0, stream>>>(x, s_ws);
  se_mlp<<<1, 256, 0, stream>>>(s_ws, w1, b1, w2, b2, gate_ws);
  se_scale<<<nbc, 256, 0, stream>>>(x, gate_ws, out);
}